// SelfAttentionPrune_87428354277762
// MI455X (gfx1250) — compile-verified
//
#include <hip/hip_runtime.h>
#include <math.h>

// ---------------------------------------------------------------------------
// SelfAttentionPrune for MI455X (gfx1250, wave32).
// fp32 end-to-end using V_WMMA_F32_16X16X4_F32.
// Flash-style (no [b,h,n,n] materialization) + double-buffered
// GLOBAL_LOAD_ASYNC_TO_LDS_B128 staging overlapped with WMMA compute.
// ---------------------------------------------------------------------------

#define B_   4
#define N_   2048
#define E_   1024
#define H_   16
#define D_   64
#define NK_  1024
#define NE3  3072

typedef float v2f __attribute__((ext_vector_type(2)));
typedef float v8f __attribute__((ext_vector_type(8)));
typedef int b128v __attribute__((vector_size(16)));   // matches builtin param type

#if defined(__has_builtin)
#if __has_builtin(__builtin_amdgcn_global_load_async_to_lds_b128) && \
    __has_builtin(__builtin_amdgcn_s_wait_asynccnt)
#define USE_ASYNC_LDS 1
#endif
#endif

// Copy 16 contiguous bytes global -> LDS. Async path uses the CDNA5
// async engine (ASYNCcnt); fallback is a sync VGPR round-trip.
__device__ __forceinline__ void cp_b128(float* lds_dst, const float* gsrc) {
#ifdef USE_ASYNC_LDS
  __builtin_amdgcn_global_load_async_to_lds_b128(
      (b128v*)gsrc,
      (__attribute__((address_space(3))) b128v*)lds_dst, 0, 0);
#else
  float4 v = *(const float4*)gsrc;
  lds_dst[0] = v.x; lds_dst[1] = v.y; lds_dst[2] = v.z; lds_dst[3] = v.w;
#endif
}

__device__ __forceinline__ void cp_wait() {
#ifdef USE_ASYNC_LDS
  __builtin_amdgcn_s_wait_asynccnt(0);
#endif
}

__device__ __forceinline__ v8f wmma4(v2f a, v2f b, v8f c) {
  // D = A(16x4) * B(4x16) + C(16x16), fp32
  return __builtin_amdgcn_wmma_f32_16x16x4_f32(
      false, a, false, b, (short)0, c, false, false);
}

// A-fragment (16x4 fp32) from row-major LDS tile [rows][stride].
// lanes 0-15: M=0..15, VGPR0=K0, VGPR1=K1; lanes 16-31: K2/K3.
// Also serves as the B^T-fragment loader when the B tile is stored
// row-major as [n][k] (lane index selects the N row of the tile).
__device__ __forceinline__ v2f ldsA(const float* base, int stride, int row0, int k0) {
  const unsigned lane = threadIdx.x & 31u;
  const unsigned l16 = lane & 15u, hf = lane >> 4;
  const float* p = base + (row0 + (int)l16) * stride + k0 + 2 * (int)hf;
  v2f a; a.x = p[0]; a.y = p[1];
  return a;
}

// ---------------------------------------------------------------------------
// K0: zero a float buffer
// ---------------------------------------------------------------------------
__global__ void k_zero(float* __restrict__ p, int nfloats) {
  int i = blockIdx.x * 256 + threadIdx.x;
  if (i < nfloats) p[i] = 0.0f;
}

// ---------------------------------------------------------------------------
// K1: QKV GEMM.  C[8192,3072] = x[8192,1024] @ Wqkv^T + bqkv
// Block tile 128x64, 8 waves (4x2), each wave 32x32 via 2x2 WMMA tiles.
// Double-buffered async LDS staging; W tiles stored row-major [col][k].
// Epilogue scatters into q/k/v workspaces laid out [b,h,n,d].
// ---------------------------------------------------------------------------
__global__ __launch_bounds__(256) void k_qkv(
    const float* __restrict__ x, const float* __restrict__ Wqkv,
    const float* __restrict__ bqkv,
    float* __restrict__ qb, float* __restrict__ kb, float* __restrict__ vb) {
  __shared__ float As[2][128 * 36];  // [m][k], stride 36 (conflict-free, 16B rows)
  __shared__ float Bs[2][64 * 36];   // [n][k], stride 36
  const int tid = threadIdx.x;
  const int wid = tid >> 5;
  const int wy = wid >> 1;           // 0..3 -> 32-row sub-block
  const int wx = wid & 1;            // 0..1 -> 32-col sub-block
  const unsigned l16 = tid & 15u, hf = (tid >> 4) & 1u;
  const int mBase = blockIdx.y * 128;
  const int nBase = blockIdx.x * 64;

  auto loadChunk = [&](int kt, int p) {
#pragma unroll
    for (int i = 0; i < 4; ++i) {                // A: 128x32 floats
      int lin = tid + i * 256;                   // 0..1023
      int row = lin >> 3, c4 = (lin & 7) << 2;
      cp_b128(&As[p][row * 36 + c4], x + (size_t)(mBase + row) * E_ + kt + c4);
    }
#pragma unroll
    for (int i = 0; i < 2; ++i) {                // B: 64x32 floats
      int lin = tid + i * 256;                   // 0..511
      int col = lin >> 3, c4 = (lin & 7) << 2;
      cp_b128(&Bs[p][col * 36 + c4],
              Wqkv + (size_t)(nBase + col) * E_ + kt + c4);
    }
  };

  v8f acc[2][2] = {};
  const int nch = E_ / 32;
  loadChunk(0, 0);
  for (int ch = 0; ch < nch; ++ch) {
    cp_wait();
    __syncthreads();
    if (ch + 1 < nch) loadChunk((ch + 1) * 32, (ch + 1) & 1);
    const float* A = As[ch & 1];
    const float* Bt = Bs[ch & 1];
#pragma unroll
    for (int kk = 0; kk < 32; kk += 4) {
      v2f a0 = ldsA(A, 36, wy * 32, kk);
      v2f a1 = ldsA(A, 36, wy * 32 + 16, kk);
      v2f b0 = ldsA(Bt, 36, wx * 32, kk);
      v2f b1 = ldsA(Bt, 36, wx * 32 + 16, kk);
      acc[0][0] = wmma4(a0, b0, acc[0][0]);
      acc[0][1] = wmma4(a0, b1, acc[0][1]);
      acc[1][0] = wmma4(a1, b0, acc[1][0]);
      acc[1][1] = wmma4(a1, b1, acc[1][1]);
    }
  }

  // epilogue: +bias, scatter to q/k/v as [b,h,n,d]
#pragma unroll
  for (int t = 0; t < 2; ++t)
#pragma unroll
    for (int u = 0; u < 2; ++u)
#pragma unroll
      for (int j = 0; j < 8; ++j) {
        int r = mBase + wy * 32 + t * 16 + (int)hf * 8 + j;   // 0..8191
        int c = nBase + wx * 32 + u * 16 + (int)l16;          // 0..3071
        float val = acc[t][u][j] + bqkv[c];
        int part = c >> 10;             // 0=q 1=k 2=v
        int ce = c & 1023;
        int head = ce >> 6, dd = ce & 63;
        int bb = r >> 11, pos = r & 2047;
        size_t off = ((size_t)(bb * H_ + head) * N_ + pos) * D_ + dd;
        float* dst = (part == 0) ? qb : ((part == 1) ? kb : vb);
        dst[off] = val;
      }
}

// ---------------------------------------------------------------------------
// K2: flash pass 1: per-row softmax stats m (max), l (sum of exp).
// One wave owns 32 query rows; loops all 2048 keys in 32-key LDS chunks,
// double-buffered async staging of K.
// ---------------------------------------------------------------------------
__global__ __launch_bounds__(256) void k_stats(
    const float* __restrict__ qb, const float* __restrict__ kb,
    float* __restrict__ mb, float* __restrict__ lb) {
  __shared__ float Ks[2][32 * 68];    // [key][d], stride 68
  __shared__ float Ssh[8][32 * 33];   // per-wave 32x32 score tile, stride 33
  const int tid = threadIdx.x, wid = tid >> 5;
  const unsigned lane = tid & 31u, l16 = lane & 15u, hf = lane >> 4;
  const int bh = blockIdx.x;                       // 0..63
  const int qBase = blockIdx.y * 256 + wid * 32;   // 0..2047
  const float* Q = qb + (size_t)bh * N_ * D_;
  const float* K = kb + (size_t)bh * N_ * D_;
  const float scale = 0.125f;  // 1/sqrt(64)

  auto loadChunk = [&](int kt, int p) {
#pragma unroll
    for (int i = 0; i < 2; ++i) {
      int lin = tid + i * 256;
      int key = lin >> 3, c4 = (lin & 7) << 2;
      cp_b128(&Ks[p][key * 68 + c4], K + (size_t)(kt + key) * D_ + c4);
    }
  };

  // Q fragments held in registers for the whole kernel
  v2f qf[2][16];
#pragma unroll
  for (int t = 0; t < 2; ++t) {
    const float* qrow = Q + (size_t)(qBase + t * 16 + (int)l16) * D_;
#pragma unroll
    for (int ks = 0; ks < 16; ++ks) {
      qf[t][ks].x = qrow[4 * ks + 2 * (int)hf];
      qf[t][ks].y = qrow[4 * ks + 2 * (int)hf + 1];
    }
  }

  float m = -3.0e38f, l = 0.0f;
  float* S = Ssh[wid];
  const int nch = N_ / 32;
  loadChunk(0, 0);
  for (int ch = 0; ch < nch; ++ch) {
    cp_wait();
    __syncthreads();
    if (ch + 1 < nch) loadChunk((ch + 1) * 32, (ch + 1) & 1);
    const float* Kt = Ks[ch & 1];

    v8f acc[2][2] = {};
#pragma unroll
    for (int ks = 0; ks < 16; ++ks) {
      int k0 = 4 * ks;
      // B = K^T: row-major [key][d] tile, lane selects key
      v2f b0 = ldsA(Kt, 68, 0, k0);
      v2f b1 = ldsA(Kt, 68, 16, k0);
      acc[0][0] = wmma4(qf[0][ks], b0, acc[0][0]);
      acc[0][1] = wmma4(qf[0][ks], b1, acc[0][1]);
      acc[1][0] = wmma4(qf[1][ks], b0, acc[1][0]);
      acc[1][1] = wmma4(qf[1][ks], b1, acc[1][1]);
    }
#pragma unroll
    for (int t = 0; t < 2; ++t)
#pragma unroll
      for (int u = 0; u < 2; ++u)
#pragma unroll
        for (int j = 0; j < 8; ++j)
          S[(t * 16 + (int)hf * 8 + j) * 33 + u * 16 + (int)l16] =
              acc[t][u][j] * scale;
    __syncthreads();

    // online softmax update; lane <-> row (stride 33 -> conflict-free)
    float chm = -3.0e38f;
#pragma unroll
    for (int c = 0; c < 32; ++c) chm = fmaxf(chm, S[lane * 33 + c]);
    float nm = fmaxf(m, chm);
    float ls = 0.0f;
#pragma unroll
    for (int c = 0; c < 32; ++c) ls += __expf(S[lane * 33 + c] - nm);
    l = l * __expf(m - nm) + ls;
    m = nm;
  }
  mb[(size_t)bh * N_ + qBase + lane] = m;
  lb[(size_t)bh * N_ + qBase + lane] = l;
}

// ---------------------------------------------------------------------------
// K3: flash pass 2: recompute scores, normalize with final (m,l),
// accumulate column sums s[b,h,k] via fp32 global atomics.
// ---------------------------------------------------------------------------
__global__ __launch_bounds__(256) void k_colsum(
    const float* __restrict__ qb, const float* __restrict__ kb,
    const float* __restrict__ mb, const float* __restrict__ lb,
    float* __restrict__ sb) {
  __shared__ float Ks[2][32 * 68];
  __shared__ float Ssh[8][32 * 33];
  __shared__ float Msh[8][32];
  __shared__ float Rsh[8][32];
  const int tid = threadIdx.x, wid = tid >> 5;
  const unsigned lane = tid & 31u, l16 = lane & 15u, hf = lane >> 4;
  const int bh = blockIdx.x;
  const int qBase = blockIdx.y * 256 + wid * 32;
  const float* Q = qb + (size_t)bh * N_ * D_;
  const float* K = kb + (size_t)bh * N_ * D_;
  const float scale = 0.125f;

  auto loadChunk = [&](int kt, int p) {
#pragma unroll
    for (int i = 0; i < 2; ++i) {
      int lin = tid + i * 256;
      int key = lin >> 3, c4 = (lin & 7) << 2;
      cp_b128(&Ks[p][key * 68 + c4], K + (size_t)(kt + key) * D_ + c4);
    }
  };

  Msh[wid][lane] = mb[(size_t)bh * N_ + qBase + lane];
  Rsh[wid][lane] = 1.0f / lb[(size_t)bh * N_ + qBase + lane];

  v2f qf[2][16];
#pragma unroll
  for (int t = 0; t < 2; ++t) {
    const float* qrow = Q + (size_t)(qBase + t * 16 + (int)l16) * D_;
#pragma unroll
    for (int ks = 0; ks < 16; ++ks) {
      qf[t][ks].x = qrow[4 * ks + 2 * (int)hf];
      qf[t][ks].y = qrow[4 * ks + 2 * (int)hf + 1];
    }
  }
  float* S = Ssh[wid];
  const int nch = N_ / 32;
  loadChunk(0, 0);
  for (int ch = 0; ch < nch; ++ch) {
    const int kt = ch * 32;
    cp_wait();
    __syncthreads();
    if (ch + 1 < nch) loadChunk((ch + 1) * 32, (ch + 1) & 1);
    const float* Kt = Ks[ch & 1];

    v8f acc[2][2] = {};
#pragma unroll
    for (int ks = 0; ks < 16; ++ks) {
      int k0 = 4 * ks;
      v2f b0 = ldsA(Kt, 68, 0, k0);
      v2f b1 = ldsA(Kt, 68, 16, k0);
      acc[0][0] = wmma4(qf[0][ks], b0, acc[0][0]);
      acc[0][1] = wmma4(qf[0][ks], b1, acc[0][1]);
      acc[1][0] = wmma4(qf[1][ks], b0, acc[1][0]);
      acc[1][1] = wmma4(qf[1][ks], b1, acc[1][1]);
    }
#pragma unroll
    for (int t = 0; t < 2; ++t)
#pragma unroll
      for (int u = 0; u < 2; ++u)
#pragma unroll
        for (int j = 0; j < 8; ++j)
          S[(t * 16 + (int)hf * 8 + j) * 33 + u * 16 + (int)l16] =
              acc[t][u][j] * scale;
    __syncthreads();

    // lane <-> column: reduce probabilities over this wave's 32 rows
    float sum = 0.0f;
#pragma unroll
    for (int r = 0; r < 32; ++r)
      sum += __expf(S[r * 33 + lane] - Msh[wid][r]) * Rsh[wid][r];
    atomicAdd(&sb[(size_t)bh * N_ + kt + lane], sum);
  }
}

// ---------------------------------------------------------------------------
// K4: exact top-1024-of-2048 per (b,h): bitonic sort in LDS by
// (value desc, index asc) to match jax.lax.top_k tie ordering.
// ---------------------------------------------------------------------------
__global__ __launch_bounds__(1024) void k_topk(
    const float* __restrict__ sb, int* __restrict__ idxb) {
  __shared__ float val[2048];
  __shared__ int ind[2048];
  const int bh = blockIdx.x, tid = threadIdx.x;
  for (int i = tid; i < 2048; i += 1024) {
    val[i] = sb[(size_t)bh * N_ + i];
    ind[i] = i;
  }
  __syncthreads();
  for (int kk = 2; kk <= 2048; kk <<= 1) {
    for (int j = kk >> 1; j > 0; j >>= 1) {
      int i = ((tid & ~(j - 1)) << 1) | (tid & (j - 1));
      int p = i | j;
      bool asc = (i & kk) != 0;  // final merge (kk=2048) is descending
      float va = val[i], vb = val[p];
      int ia = ind[i], ib = ind[p];
      bool aBefore = (va > vb) || (va == vb && ia < ib);  // desc predicate
      bool doSwap = asc ? aBefore : !aBefore;
      if (doSwap) { val[i] = vb; val[p] = va; ind[i] = ib; ind[p] = ia; }
      __syncthreads();
    }
  }
  idxb[(size_t)bh * NK_ + tid] = ind[tid];
}

// ---------------------------------------------------------------------------
// K5: attention output for the 1024 selected query rows per (b,h).
// Gather Q rows by idx, QK^T (WMMA), p = exp(s-m)/l, then P*V (WMMA).
// Double-buffered async staging of K and V.
// ---------------------------------------------------------------------------
__global__ __launch_bounds__(256) void k_attnout(
    const float* __restrict__ qb, const float* __restrict__ kbuf,
    const float* __restrict__ vbuf, const float* __restrict__ mb,
    const float* __restrict__ lb, const int* __restrict__ idxb,
    float* __restrict__ ob) {
  __shared__ float Ks[2][32 * 68];
  __shared__ float Vs[2][32 * 68];
  __shared__ float Ssh[8][32 * 33];
  const int tid = threadIdx.x, wid = tid >> 5;
  const unsigned lane = tid & 31u, l16 = lane & 15u, hf = lane >> 4;
  const int bh = blockIdx.x, bb = bh >> 4, hh = bh & 15;
  const int selBase = blockIdx.y * 256 + wid * 32;  // position in idx list
  const float* Q = qb + (size_t)bh * N_ * D_;
  const float* K = kbuf + (size_t)bh * N_ * D_;
  const float* V = vbuf + (size_t)bh * N_ * D_;
  const float scale = 0.125f;

  auto loadChunk = [&](int kt, int p) {
#pragma unroll
    for (int i = 0; i < 2; ++i) {
      int lin = tid + i * 256;
      int key = lin >> 3, c4 = (lin & 7) << 2;
      cp_b128(&Ks[p][key * 68 + c4], K + (size_t)(kt + key) * D_ + c4);
      cp_b128(&Vs[p][key * 68 + c4], V + (size_t)(kt + key) * D_ + c4);
    }
  };

  const int myrow = idxb[(size_t)bh * NK_ + selBase + lane];
  const float mrow = mb[(size_t)bh * N_ + myrow];
  const float rinv = 1.0f / lb[(size_t)bh * N_ + myrow];

  v2f qf[2][16];
#pragma unroll
  for (int t = 0; t < 2; ++t) {
    int qi = idxb[(size_t)bh * NK_ + selBase + t * 16 + (int)l16];
    const float* qrow = Q + (size_t)qi * D_;
#pragma unroll
    for (int ks = 0; ks < 16; ++ks) {
      qf[t][ks].x = qrow[4 * ks + 2 * (int)hf];
      qf[t][ks].y = qrow[4 * ks + 2 * (int)hf + 1];
    }
  }

  v8f oacc[2][4] = {};
  float* S = Ssh[wid];
  const int nch = N_ / 32;
  loadChunk(0, 0);
  for (int ch = 0; ch < nch; ++ch) {
    cp_wait();
    __syncthreads();
    if (ch + 1 < nch) loadChunk((ch + 1) * 32, (ch + 1) & 1);
    const float* Kt = Ks[ch & 1];
    const float* Vt = Vs[ch & 1];

    v8f acc[2][2] = {};
#pragma unroll
    for (int ks = 0; ks < 16; ++ks) {
      int k0 = 4 * ks;
      v2f b0 = ldsA(Kt, 68, 0, k0);
      v2f b1 = ldsA(Kt, 68, 16, k0);
      acc[0][0] = wmma4(qf[0][ks], b0, acc[0][0]);
      acc[0][1] = wmma4(qf[0][ks], b1, acc[0][1]);
      acc[1][0] = wmma4(qf[1][ks], b0, acc[1][0]);
      acc[1][1] = wmma4(qf[1][ks], b1, acc[1][1]);
    }
#pragma unroll
    for (int t = 0; t < 2; ++t)
#pragma unroll
      for (int u = 0; u < 2; ++u)
#pragma unroll
        for (int j = 0; j < 8; ++j)
          S[(t * 16 + (int)hf * 8 + j) * 33 + u * 16 + (int)l16] =
              acc[t][u][j] * scale;
    __syncthreads();

    // normalize in place; lane <-> row
#pragma unroll
    for (int c = 0; c < 32; ++c)
      S[lane * 33 + c] = __expf(S[lane * 33 + c] - mrow) * rinv;
    __syncthreads();

    // o += P(32x32) * V(32x64); P re-read from LDS in A layout,
    // V-fragment: B[key][dcol] pattern on the [key][d] tile
#pragma unroll
    for (int ks2 = 0; ks2 < 8; ++ks2) {
      int k0 = 4 * ks2;
      v2f a0 = ldsA(S, 33, 0, k0);
      v2f a1 = ldsA(S, 33, 16, k0);
#pragma unroll
      for (int u = 0; u < 4; ++u) {
        v2f b;
        b.x = Vt[(k0 + 2 * (int)hf) * 68 + u * 16 + (int)l16];
        b.y = Vt[(k0 + 2 * (int)hf + 1) * 68 + u * 16 + (int)l16];
        oacc[0][u] = wmma4(a0, b, oacc[0][u]);
        oacc[1][u] = wmma4(a1, b, oacc[1][u]);
      }
    }
  }

  // write o[b, sel_pos, h*64+dd]
#pragma unroll
  for (int t = 0; t < 2; ++t)
#pragma unroll
    for (int u = 0; u < 4; ++u)
#pragma unroll
      for (int j = 0; j < 8; ++j) {
        int r = selBase + t * 16 + (int)hf * 8 + j;
        int c = hh * D_ + u * 16 + (int)l16;
        ob[((size_t)bb * NK_ + r) * E_ + c] = oacc[t][u][j];
      }
}

// ---------------------------------------------------------------------------
// K6: out = o[4096,1024] @ Wout^T + bout + per-head gathered residual
// ---------------------------------------------------------------------------
__global__ __launch_bounds__(256) void k_outproj(
    const float* __restrict__ ob, const float* __restrict__ Wout,
    const float* __restrict__ bout, const float* __restrict__ x,
    const int* __restrict__ idxb, float* __restrict__ out) {
  __shared__ float As[2][128 * 36];
  __shared__ float Bs[2][64 * 36];
  const int tid = threadIdx.x;
  const int wid = tid >> 5;
  const int wy = wid >> 1, wx = wid & 1;
  const unsigned l16 = tid & 15u, hf = (tid >> 4) & 1u;
  const int mBase = blockIdx.y * 128;   // over 4096 rows
  const int nBase = blockIdx.x * 64;    // over 1024 cols

  auto loadChunk = [&](int kt, int p) {
#pragma unroll
    for (int i = 0; i < 4; ++i) {
      int lin = tid + i * 256;
      int row = lin >> 3, c4 = (lin & 7) << 2;
      cp_b128(&As[p][row * 36 + c4], ob + (size_t)(mBase + row) * E_ + kt + c4);
    }
#pragma unroll
    for (int i = 0; i < 2; ++i) {
      int lin = tid + i * 256;
      int col = lin >> 3, c4 = (lin & 7) << 2;
      cp_b128(&Bs[p][col * 36 + c4],
              Wout + (size_t)(nBase + col) * E_ + kt + c4);
    }
  };

  v8f acc[2][2] = {};
  const int nch = E_ / 32;
  loadChunk(0, 0);
  for (int ch = 0; ch < nch; ++ch) {
    cp_wait();
    __syncthreads();
    if (ch + 1 < nch) loadChunk((ch + 1) * 32, (ch + 1) & 1);
    const float* A = As[ch & 1];
    const float* Bt = Bs[ch & 1];
#pragma unroll
    for (int kk = 0; kk < 32; kk += 4) {
      v2f a0 = ldsA(A, 36, wy * 32, kk);
      v2f a1 = ldsA(A, 36, wy * 32 + 16, kk);
      v2f b0 = ldsA(Bt, 36, wx * 32, kk);
      v2f b1 = ldsA(Bt, 36, wx * 32 + 16, kk);
      acc[0][0] = wmma4(a0, b0, acc[0][0]);
      acc[0][1] = wmma4(a0, b1, acc[0][1]);
      acc[1][0] = wmma4(a1, b0, acc[1][0]);
      acc[1][1] = wmma4(a1, b1, acc[1][1]);
    }
  }

#pragma unroll
  for (int t = 0; t < 2; ++t)
#pragma unroll
    for (int u = 0; u < 2; ++u)
#pragma unroll
      for (int j = 0; j < 8; ++j) {
        int r = mBase + wy * 32 + t * 16 + (int)hf * 8 + j;  // 0..4095
        int c = nBase + wx * 32 + u * 16 + (int)l16;         // 0..1023
        int bb = r >> 10, i = r & 1023;
        int head = c >> 6;
        int key = idxb[(size_t)(bb * H_ + head) * NK_ + i];
        float res = x[((size_t)bb * N_ + key) * E_ + c];
        out[(size_t)r * E_ + c] = acc[t][u][j] + bout[c] + res;
      }
}

// ---------------------------------------------------------------------------
// Host-side launch
// ---------------------------------------------------------------------------
extern "C" void kernel_launch(void* const* d_in, const int* in_sizes, int n_in,
                              void* d_out, int out_size, void* d_ws,
                              size_t ws_size, hipStream_t stream) {
  const float* x    = (const float*)d_in[0];
  const float* Wqkv = (const float*)d_in[1];
  const float* bqkv = (const float*)d_in[2];
  const float* Wout = (const float*)d_in[3];
  const float* bout = (const float*)d_in[4];
  float* out = (float*)d_out;

  // workspace layout (floats); total ~119 MB
  float* ws = (float*)d_ws;
  const size_t qkvStride = (size_t)B_ * H_ * N_ * D_;  // 8M floats
  float* qb   = ws;
  float* kb   = qb + qkvStride;
  float* vb   = kb + qkvStride;
  float* obuf = vb + qkvStride;                        // 4M floats
  float* mb   = obuf + (size_t)B_ * NK_ * E_;          // 128K
  float* lb   = mb + (size_t)B_ * H_ * N_;
  float* sbuf = lb + (size_t)B_ * H_ * N_;
  int*   idxb = (int*)(sbuf + (size_t)B_ * H_ * N_);

  const int nS = B_ * H_ * N_;  // 131072
  k_zero<<<(nS + 255) / 256, 256, 0, stream>>>(sbuf, nS);

  k_qkv<<<dim3(NE3 / 64, (B_ * N_) / 128), 256, 0, stream>>>(
      x, Wqkv, bqkv, qb, kb, vb);

  k_stats<<<dim3(B_ * H_, N_ / 256), 256, 0, stream>>>(qb, kb, mb, lb);

  k_colsum<<<dim3(B_ * H_, N_ / 256), 256, 0, stream>>>(qb, kb, mb, lb, sbuf);

  k_topk<<<B_ * H_, 1024, 0, stream>>>(sbuf, idxb);

  k_attnout<<<dim3(B_ * H_, NK_ / 256), 256, 0, stream>>>(
      qb, kb, vb, mb, lb, idxb, obuf);

  k_outproj<<<dim3(E_ / 64, (B_ * NK_) / 128), 256, 0, stream>>>(
      obuf, Wout, bout, x, idxb, out);
}